// PartialPooling2d_23639499997404
// MI455X (gfx1250) — compile-verified
//
#include <hip/hip_runtime.h>
#include <stdint.h>

// (16, 64, 512, 512) f32 -> same-shape output; pooled 256x256 corner + zeros.
#define W_DIM    512
#define H2_DIM   256
#define PLANES   1024      // 16 * 64
#define RPB      4         // output rows per block
#define TPB      256       // 8 wave32 per block

typedef float f4  __attribute__((ext_vector_type(4)));
typedef int   v4i __attribute__((ext_vector_type(4)));

// gfx1250 async LDS<->global store path (builtin confirmed present by round-1 probe;
// signature: (v4i addrspace(1)* gdst, v4i addrspace(3)* lds_src, imm offset, imm cpol)).
#if defined(__HIP_DEVICE_COMPILE__) && __has_builtin(__builtin_amdgcn_global_store_async_from_lds_b128)
#  define HAVE_ASYNC_ZERO 1
#else
#  define HAVE_ASYNC_ZERO 0
#endif

#if HAVE_ASYNC_ZERO
typedef __attribute__((address_space(1))) v4i gv4i;   // global
typedef __attribute__((address_space(3))) v4i lv4i;   // LDS
#endif

__global__ __launch_bounds__(TPB) void pool_ckbd_kernel(const float* __restrict__ in,
                                                        float* __restrict__ out) {
    const uint32_t bid  = blockIdx.x;
    const uint32_t p    = bid >> 6;          // plane 0..1023
    const uint32_t rg   = bid & 63u;         // row-group 0..63
    const uint32_t tid  = threadIdx.x;
    const uint32_t r    = tid >> 6;          // 0..3   row within group
    const uint32_t lane = tid & 63u;         // 0..63  64 lanes cover 256 cols (f4 each)
    const uint32_t i    = (rg << 2) | r;     // output row 0..255

    const size_t planeOff = (size_t)p * (size_t)(W_DIM * W_DIM);

    // Input rows 2i and 2i+1; lane covers input cols [8*lane, 8*lane+7].
    const f4* in0 = (const f4*)(in + planeOff + (size_t)(2u * i)      * W_DIM) + 2u * lane;
    const f4* in1 = (const f4*)(in + planeOff + (size_t)(2u * i + 1u) * W_DIM) + 2u * lane;

    // Streamed once: non-temporal loads keep the 192MB L2 clean.
    f4 x0 = __builtin_nontemporal_load(in0);
    f4 x1 = __builtin_nontemporal_load(in0 + 1);
    f4 y0 = __builtin_nontemporal_load(in1);
    f4 y1 = __builtin_nontemporal_load(in1 + 1);

    // Output cols j = 4*lane + k, k = 0..3; window = cols (2j, 2j+1) of rows (2i, 2i+1).
    float m0 = fmaxf(fmaxf(x0.x, x0.y), fmaxf(y0.x, y0.y));
    float m1 = fmaxf(fmaxf(x0.z, x0.w), fmaxf(y0.z, y0.w));
    float m2 = fmaxf(fmaxf(x1.x, x1.y), fmaxf(y1.x, y1.y));
    float m3 = fmaxf(fmaxf(x1.z, x1.w), fmaxf(y1.z, y1.w));

    float a0 = (x0.x + x0.y + y0.x + y0.y) * 0.25f;
    float a1 = (x0.z + x0.w + y0.z + y0.w) * 0.25f;
    float a2 = (x1.x + x1.y + y1.x + y1.y) * 0.25f;
    float a3 = (x1.z + x1.w + y1.z + y1.w) * 0.25f;

    // use_max = ((i + j) % 2 == 0); 4*lane is even, so parity depends only on (i + k).
    const bool ieven = (i & 1u) == 0u;
    f4 res;
    res.x = ieven ? m0 : a0;   // k = 0
    res.y = ieven ? a1 : m1;   // k = 1
    res.z = ieven ? m2 : a2;   // k = 2
    res.w = ieven ? a3 : m3;   // k = 3

    f4* orow = (f4*)(out + planeOff + (size_t)i * W_DIM);
    __builtin_nontemporal_store(res, orow + lane);

    // Zero regions owned by this block:
    //   row i,      cols 256..511  (64 f4, one per lane)
    //   row 256+i,  cols 0..511    (128 f4, two per lane)
    f4* zrow = (f4*)(out + planeOff + (size_t)(H2_DIM + i) * W_DIM);

#if HAVE_ASYNC_ZERO
    // Zeros have no data dependence: DMA them from a 16B zeroed LDS block via
    // the gfx1250 async store path (ASYNCcnt), freeing the VALU/VGPR pipe.
    __shared__ __align__(16) float zsh[4];
    if (tid < 4u) zsh[tid] = 0.0f;
    __syncthreads();

    // Flat shared address: low 32 bits are the LDS byte offset (ISA aperture rule);
    // int->pointer casts sidestep clang's ban on cross-address-space pointer casts.
    lv4i* lsrc = (lv4i*)(uint32_t)(uintptr_t)&zsh[0];

    __builtin_amdgcn_global_store_async_from_lds_b128((gv4i*)(uintptr_t)(orow + 64 + lane), lsrc, 0, 0);
    __builtin_amdgcn_global_store_async_from_lds_b128((gv4i*)(uintptr_t)(zrow + lane),      lsrc, 0, 0);
    __builtin_amdgcn_global_store_async_from_lds_b128((gv4i*)(uintptr_t)(zrow + 64 + lane), lsrc, 0, 0);
#  if __has_builtin(__builtin_amdgcn_s_wait_asynccnt)
    __builtin_amdgcn_s_wait_asynccnt(0);
#  endif
    // (S_ENDPGM performs an implicit wait-idle, so completion is guaranteed regardless.)
#else
    f4 z = {0.0f, 0.0f, 0.0f, 0.0f};
    __builtin_nontemporal_store(z, orow + 64 + lane);
    __builtin_nontemporal_store(z, zrow + lane);
    __builtin_nontemporal_store(z, zrow + 64 + lane);
#endif
}

extern "C" void kernel_launch(void* const* d_in, const int* in_sizes, int n_in,
                              void* d_out, int out_size, void* d_ws, size_t ws_size,
                              hipStream_t stream) {
    (void)in_sizes; (void)n_in; (void)out_size; (void)d_ws; (void)ws_size;
    const float* x   = (const float*)d_in[0];
    float*       out = (float*)d_out;

    dim3 grid(PLANES * (H2_DIM / RPB));   // 1024 * 64 = 65536 blocks
    dim3 block(TPB);                      // 256 threads = 8 wave32
    hipLaunchKernelGGL(pool_ckbd_kernel, grid, block, 0, stream, x, out);
}